// DiffConvLayer_11828339933447
// MI455X (gfx1250) — compile-verified
//
#include <hip/hip_runtime.h>
#include <hip/hip_bf16.h>

typedef float v2f __attribute__((ext_vector_type(2)));
typedef float v8f __attribute__((ext_vector_type(8)));

#define N_NODES 20000
#define E_EDGES 640000
#define FDIM    128

// Fused dual GEMM with fp32 WMMA (V_WMMA_F32_16X16X4_F32).
// Per wave: one 16x16 output tile of BOTH n0 = x@W0 and n1 = x@W1.
// Epilogue: n0s = (merger0/3)*n0  (pre-scaled for graph-0 scatter)
//           n1  = x@W1            (unscaled, gathered with merger1/3 later)
//           out = (merger2/3)*n1  (the reused-projection term f2)
__global__ __launch_bounds__(128) void diffconv_gemm_wmma(
    const float* __restrict__ x,
    const float* __restrict__ W0,
    const float* __restrict__ W1,
    const float* __restrict__ merger,
    float* __restrict__ n0s,
    float* __restrict__ n1,
    float* __restrict__ out)
{
    const int wave = blockIdx.x * (blockDim.x >> 5) + (threadIdx.x >> 5);
    const int lane = threadIdx.x & 31;

    const int numTilesN = FDIM / 16;            // 8
    const int tileM = wave / numTilesN;         // 0..1249
    const int tileN = wave % numTilesN;
    if (tileM >= N_NODES / 16) return;

    const int mn    = lane & 15;                // M index (A) / N index (B,C)
    const int khalf = lane >> 4;                // selects K pair {0,1} vs {2,3}

    const float s0 = merger[0] * (1.0f / 3.0f);
    const float s2 = merger[2] * (1.0f / 3.0f);

    const float* __restrict__ xrow = x + (size_t)(tileM * 16 + mn) * FDIM;
    const int ncol = tileN * 16 + mn;

    v8f acc0 = {};
    v8f acc1 = {};

    for (int k = 0; k < FDIM; k += 4) {
        const int ka = k + 2 * khalf;
        // A fragment: lane = M + 16*khalf, VGPR0/1 = K = ka, ka+1
        v2f a;
        a.x = xrow[ka];
        a.y = xrow[ka + 1];
        // B fragments: lane = N + 16*khalf, VGPR0/1 = K = ka, ka+1
        v2f b0;
        b0.x = W0[(size_t)ka * FDIM + ncol];
        b0.y = W0[(size_t)(ka + 1) * FDIM + ncol];
        v2f b1;
        b1.x = W1[(size_t)ka * FDIM + ncol];
        b1.y = W1[(size_t)(ka + 1) * FDIM + ncol];

        acc0 = __builtin_amdgcn_wmma_f32_16x16x4_f32(
            false, a, false, b0, (short)0, acc0, false, false);
        acc1 = __builtin_amdgcn_wmma_f32_16x16x4_f32(
            false, a, false, b1, (short)0, acc1, false, false);
    }

    // C/D layout: VGPR r -> M = r + 8*khalf (within tile), N = lane&15
    const int mBase = tileM * 16 + 8 * khalf;
    #pragma unroll
    for (int r = 0; r < 8; ++r) {
        const size_t idx = (size_t)(mBase + r) * FDIM + (size_t)(tileN * 16 + mn);
        n0s[idx] = acc0[r] * s0;
        n1[idx]  = acc1[r];
        out[idx] = acc1[r] * s2;
    }
}

// Fused scatter over both graphs: one wave32 per edge; lane l owns features
// [4l, 4l+4). Gather float4 from the (L2-resident) projected matrix, scale by
// the edge weight (merger factor pre-folded), atomically accumulate into out.
__global__ __launch_bounds__(256) void diffconv_scatter(
    const int*   __restrict__ src0, const int* __restrict__ dst0,
    const float* __restrict__ w0,
    const int*   __restrict__ src1, const int* __restrict__ dst1,
    const float* __restrict__ w1,
    const float* __restrict__ n0s,  const float* __restrict__ n1,
    const float* __restrict__ merger,
    float* __restrict__ out)
{
    const long e = (long)blockIdx.x * (blockDim.x >> 5) + (threadIdx.x >> 5);
    if (e >= 2L * E_EDGES) return;
    const int lane = threadIdx.x & 31;

    int src, dst;
    float coef;
    const float* __restrict__ nsrc;
    if (e < E_EDGES) {
        src  = src0[e];
        dst  = dst0[e];
        coef = w0[e];            // merger0/3 already folded into n0s
        nsrc = n0s;
    } else {
        const long e1 = e - E_EDGES;
        src  = src1[e1];
        dst  = dst1[e1];
        coef = w1[e1] * (merger[1] * (1.0f / 3.0f));
        nsrc = n1;
    }

    const float4 v =
        *reinterpret_cast<const float4*>(nsrc + (size_t)src * FDIM + 4 * lane);
    float* o = out + (size_t)dst * FDIM + 4 * lane;

    __hip_atomic_fetch_add(o + 0, v.x * coef, __ATOMIC_RELAXED, __HIP_MEMORY_SCOPE_AGENT);
    __hip_atomic_fetch_add(o + 1, v.y * coef, __ATOMIC_RELAXED, __HIP_MEMORY_SCOPE_AGENT);
    __hip_atomic_fetch_add(o + 2, v.z * coef, __ATOMIC_RELAXED, __HIP_MEMORY_SCOPE_AGENT);
    __hip_atomic_fetch_add(o + 3, v.w * coef, __ATOMIC_RELAXED, __HIP_MEMORY_SCOPE_AGENT);
}

extern "C" void kernel_launch(void* const* d_in, const int* in_sizes, int n_in,
                              void* d_out, int out_size, void* d_ws, size_t ws_size,
                              hipStream_t stream) {
    const float* x      = (const float*)d_in[0];
    const int*   src0   = (const int*)  d_in[1];
    const int*   dst0   = (const int*)  d_in[2];
    const float* w0     = (const float*)d_in[3];
    const int*   src1   = (const int*)  d_in[4];
    const int*   dst1   = (const int*)  d_in[5];
    const float* w1     = (const float*)d_in[6];
    const float* W0     = (const float*)d_in[7];
    const float* W1     = (const float*)d_in[8];
    const float* merger = (const float*)d_in[9];

    float* out = (float*)d_out;
    float* n0s = (float*)d_ws;                       // [N, F] pre-scaled proj 0
    float* n1  = n0s + (size_t)N_NODES * FDIM;       // [N, F] proj 1

    // GEMM phase: 1250 M-tiles x 8 N-tiles = 10000 waves; 4 waves / 128-thread block.
    const int numWaves  = (N_NODES / 16) * (FDIM / 16);
    const int gemmBlocks = (numWaves + 3) / 4;
    diffconv_gemm_wmma<<<gemmBlocks, 128, 0, stream>>>(
        x, W0, W1, merger, n0s, n1, out);

    // Scatter phase: 2E edges, 1 wave per edge, 8 waves / 256-thread block.
    const long totalEdges   = 2L * E_EDGES;
    const int  scatterBlocks = (int)((totalEdges + 7) / 8);
    diffconv_scatter<<<scatterBlocks, 256, 0, stream>>>(
        src0, dst0, w0, src1, dst1, w1, n0s, n1, merger, out);
}